// MessagePassing_15015205667407
// MI455X (gfx1250) — compile-verified
//
#include <hip/hip_runtime.h>
#include <cstdint>

#define H_ 256
#define W_ 512
#define C_ 64
#define HW_ (H_ * W_)
#define K_ 8          // async prefetch depth (scan positions ahead)
#define NSLOT 9       // K_+1 ring slots so an issue never targets the slot being read

// Async copy 3 dwords (prob, mA, mB for one channel) into an LDS ring slot.
__device__ __forceinline__ void async3(unsigned lds, const float* p, const float* a, const float* b) {
  asm volatile("global_load_async_to_lds_b32 %0, %1, off"
               :: "v"(lds), "v"((unsigned long long)(uintptr_t)p) : "memory");
  asm volatile("global_load_async_to_lds_b32 %0, %1, off"
               :: "v"(lds + 4u), "v"((unsigned long long)(uintptr_t)a) : "memory");
  asm volatile("global_load_async_to_lds_b32 %0, %1, off"
               :: "v"(lds + 8u), "v"((unsigned long long)(uintptr_t)b) : "memory");
}

__global__ void __launch_bounds__(64) sweep_kernel(
    const float* __restrict__ prob,
    const float* __restrict__ ew,
    const float* __restrict__ m_in,
    const float* __restrict__ jmp,
    const int*   __restrict__ spp,
    float* __restrict__ m_out)
{
  __shared__ float4 ring[NSLOT * 64];     // per-channel {p, mA, mB, pad}
  __shared__ __align__(16) float tmpbuf[64];
  __shared__ float red[2];

  const int t = threadIdx.x;
  const int b = blockIdx.x;
  // Direction + scanline decode (branch-free-ish; used only outside the hot loop).
  const int d    = (b < 2 * H_) ? ((b < H_) ? 0 : 1) : ((b < 2 * H_ + W_) ? 2 : 3);
  const int line = b - ((d == 0) ? 0 : (d == 1) ? H_ : (d == 2) ? 2 * H_ : 2 * H_ + W_);
  const int L      = (d < 2) ? W_ : H_;
  const int stride = (d == 0) ? 1 : (d == 1) ? -1 : (d == 2) ? W_ : -W_;
  const int start  = (d == 0) ? line * W_
                   : (d == 1) ? line * W_ + (W_ - 1)
                   : (d == 2) ? line
                   :            (H_ - 1) * W_ + line;
  const int strideC = stride * C_;

  const int sp = *spp;
  // Pairwise-cost column for this dst channel t (clipped at DELTA=2.0).
  // d0: J0[s][t]; d1: J0^T; d2: J1[s][t]; d3: J1^T. single_pw: J0 for all dirs.
  const int off  = (!sp && d >= 2) ? (C_ * C_) : 0;
  const bool tr  = (!sp) && (d & 1);
  const int sStr = tr ? 1 : C_;
  const int tStr = tr ? C_ : 1;
  float pwc[C_];
  #pragma unroll
  for (int s = 0; s < C_; ++s)
    pwc[s] = fminf(jmp[off + s * sStr + t * tStr], 2.0f);

  const float* mA  = m_in  + (size_t)((d < 2) ? 2 : 0) * HW_ * C_;
  const float* mB  = m_in  + (size_t)((d < 2) ? 3 : 1) * HW_ * C_;
  const float* mD  = m_in  + (size_t)d * HW_ * C_;
  float*       mO  = m_out + (size_t)d * HW_ * C_;
  const float* ewd = ew    + (size_t)d * HW_;

  const unsigned ringBase = (unsigned)(unsigned long long)(uintptr_t)&ring[0];

  // Prologue: async-prefetch unary inputs for scan positions 0..K_-1 (L >= K_+2 always).
  {
    const float* pP = prob + (size_t)start * C_ + t;
    const float* pA = mA   + (size_t)start * C_ + t;
    const float* pB = mB   + (size_t)start * C_ + t;
    #pragma unroll
    for (int j = 0; j < K_; ++j) {
      unsigned lds = ringBase + (unsigned)((j * 64 + t) << 4);
      async3(lds, pP, pA, pB);
      pP += strideC; pA += strideC; pB += strideC;
    }
  }

  // Boundary (scan index 0): message passes through unchanged.
  float m = mD[(size_t)start * C_ + t];
  mO[(size_t)start * C_ + t] = m;

  // Incremental pointers for the hot loop.
  const float* wp  = ewd + start;                       // weight at current pos i
  float*       mOp = mO + (size_t)start * C_ + t;       // output at current pos i
  // Prefetch pointers sit at scan position jn (clamped at L-2).
  int jn = K_;
  const float* pPf = prob + (size_t)(start + K_ * stride) * C_ + t;
  const float* aPf = mA   + (size_t)(start + K_ * stride) * C_ + t;
  const float* bPf = mB   + (size_t)(start + K_ * stride) * C_ + t;

  int slotC = 0;    // slot holding position j = i-1
  int slotI = K_;   // slot for position j+K_
  for (int i = 1; i < L; ++i) {
    // Oldest in-flight position (j=i-1) must be done: K_ positions x 3 ops in flight.
    asm volatile("s_wait_asynccnt %0" :: "i"(3 * (K_ - 1)) : "memory");
    // Consume unary inputs at scan position i-1 for this channel.
    float4 v = ring[slotC * 64 + t];
    // Refill the pipeline K_ positions ahead. Near the tail the pointers stop
    // advancing (repeated dummy reloads of position L-2) so the outstanding
    // count stays uniform at 3*K_ and the immediate wait above stays valid.
    async3(ringBase + (unsigned)((slotI * 64 + t) << 4), pPf, aPf, bPf);
    if (jn < L - 2) { ++jn; pPf += strideC; aPf += strideC; bPf += strideC; }

    float u = v.y + v.z - v.x;          // unary = cost + mA + mB = -p + a + b
    tmpbuf[t] = u + m;
    __syncthreads();

    wp += stride; mOp += strideC;
    const float w = *wp;                // uniform across block -> scalar load
    float acc = 3.0e38f;
    const float4* t4 = (const float4*)tmpbuf;
    #pragma unroll
    for (int sb = 0; sb < 16; ++sb) {   // 64x min-plus: min(acc, tmp[s] + w*pw[s][t])
      float4 tv = t4[sb];
      acc = fminf(acc, fmaf(w, pwc[4 * sb + 0], tv.x));
      acc = fminf(acc, fmaf(w, pwc[4 * sb + 1], tv.y));
      acc = fminf(acc, fmaf(w, pwc[4 * sb + 2], tv.z));
      acc = fminf(acc, fmaf(w, pwc[4 * sb + 3], tv.w));
    }
    // Normalize: subtract min over the 64 channels (2 waves).
    float r = acc;
    #pragma unroll
    for (int msk = 1; msk < 32; msk <<= 1)
      r = fminf(r, __shfl_xor(r, msk, 32));
    if ((t & 31) == 0) red[t >> 5] = r;
    __syncthreads();
    float mu = fminf(red[0], red[1]);
    m = acc - mu;
    *mOp = m;

    slotC = (slotC + 1 == NSLOT) ? 0 : slotC + 1;
    slotI = (slotI + 1 == NSLOT) ? 0 : slotI + 1;
  }
}

// beliefs = sum_d m[d] + cost; out = softmax(-beliefs) = softmax(prob - sum m).
// One wave32 per pixel, two channels per lane, shuffle-only reductions.
__global__ void __launch_bounds__(256) belief_softmax_kernel(
    const float* __restrict__ prob,
    const float* __restrict__ m,
    float* __restrict__ out)
{
  const int wid  = threadIdx.x >> 5;
  const int lane = threadIdx.x & 31;
  const int pix  = blockIdx.x * 8 + wid;
  const size_t base = (size_t)pix * C_;
  const size_t P = (size_t)HW_ * C_;
  const size_t i0 = base + lane, i1 = base + lane + 32;
  float v0 = prob[i0] - (m[i0] + m[P + i0] + m[2 * P + i0] + m[3 * P + i0]);
  float v1 = prob[i1] - (m[i1] + m[P + i1] + m[2 * P + i1] + m[3 * P + i1]);
  float mx = fmaxf(v0, v1);
  #pragma unroll
  for (int msk = 1; msk < 32; msk <<= 1) mx = fmaxf(mx, __shfl_xor(mx, msk, 32));
  float e0 = __expf(v0 - mx), e1 = __expf(v1 - mx);
  float s = e0 + e1;
  #pragma unroll
  for (int msk = 1; msk < 32; msk <<= 1) s += __shfl_xor(s, msk, 32);
  out[i0] = e0 / s;
  out[i1] = e1 / s;
}

extern "C" void kernel_launch(void* const* d_in, const int* in_sizes, int n_in,
                              void* d_out, int out_size, void* d_ws, size_t ws_size,
                              hipStream_t stream) {
  (void)in_sizes; (void)n_in; (void)out_size; (void)ws_size;
  const float* prob = (const float*)d_in[0];   // [1,256,512,64]
  const float* ew   = (const float*)d_in[1];   // [1,4,256,512]
  const float* msg  = (const float*)d_in[2];   // [1,4,256,512,64]
  const float* jmp  = (const float*)d_in[3];   // [1,2,64,64]
  const int*   sp   = (const int*)d_in[4];     // scalar flag
  float* A = (float*)d_ws;                     // iter-1 messages (134 MB)
  float* B = A + (size_t)4 * HW_ * C_;         // iter-2 messages (134 MB)
  float* out = (float*)d_out;

  dim3 grid(2 * H_ + 2 * W_), blk(64);
  // MAX_ITER = 2: all 4 directions of one iteration read only old messages -> one launch each.
  sweep_kernel<<<grid, blk, 0, stream>>>(prob, ew, msg, jmp, sp, A);
  sweep_kernel<<<grid, blk, 0, stream>>>(prob, ew, A,   jmp, sp, B);
  belief_softmax_kernel<<<dim3(HW_ / 8), dim3(256), 0, stream>>>(prob, B, out);
}